// PointPillarScatter_33019708571995
// MI455X (gfx1250) — compile-verified
//
#include <hip/hip_runtime.h>

// ---------------------------------------------------------------------------
// PointPillarScatter for MI455X (gfx1250, wave32):
//   h = pf @ W + b; h = BN(h); h = relu(h); scatter to [B, C, NY, NX]
// Stage 1: nontemporal zero-fill of the 438 MB output (store-bandwidth bound).
// Stage 2: fused WMMA-f32 GEMM + BN + ReLU + scatter, with weights staged in
//          LDS pre-swizzled into per-lane B-fragment order so each fragment is
//          one conflict-free ds_load_b64 (no VALU repacking before WMMA).
// ---------------------------------------------------------------------------

#define C_DIM   64
#define NX_DIM  432
#define NY_DIM  496
#define B_DIM   8
#define P_TOTAL (8 * 16384)
#define BN_EPS  1e-5f

typedef __attribute__((ext_vector_type(2))) float v2f;
typedef __attribute__((ext_vector_type(4))) float v4f;
typedef __attribute__((ext_vector_type(8))) float v8f;

// ---------------------------------------------------------------------------
// Zero-fill kernel: grid-stride float4 nontemporal stores.
// ---------------------------------------------------------------------------
__global__ __launch_bounds__(256) void pps_zero_kernel(v4f* __restrict__ out,
                                                       unsigned int n4) {
    unsigned int i = blockIdx.x * blockDim.x + threadIdx.x;
    unsigned int stride = gridDim.x * blockDim.x;
    v4f z = {0.f, 0.f, 0.f, 0.f};
    for (; i < n4; i += stride) {
        __builtin_nontemporal_store(z, out + i);
    }
}

// ---------------------------------------------------------------------------
// Fused GEMM (V_WMMA_F32_16X16X4_F32) + BatchNorm + ReLU + scatter.
// One wave computes a 16(pillar) x 64(channel) tile:
//   - A frag (16x4 f32): lanes 0-15 hold K={k,k+1}, lanes 16-31 hold K={k+2,k+3}
//   - B frag (4x16 f32): VGPR0 = row K (k / k+2), VGPR1 = row K+1, N = lane&15
//   - C/D (16x16 f32):   VGPR v, lanes 0-15 -> M=v, lanes 16-31 -> M=v+8
// 4 waves per block. LDS holds B fragments pre-swizzled:
//   lds_bf[(ks*4 + nt)*32 + lane] = { w[(4ks+koff)*64 + n], w[(4ks+koff+1)*64 + n] }
//   with koff = (lane>=16)*2, n = nt*16 + (lane&15).
// ---------------------------------------------------------------------------
__global__ __launch_bounds__(128) void pps_gemm_scatter_kernel(
    const float* __restrict__ pf,      // [P, 64]
    const int*   __restrict__ coords,  // [P, 5] : b, x, y, z, flag
    const float* __restrict__ w,       // [64, 64] (K-major: w[k*64 + n])
    const float* __restrict__ bias,    // [64]
    const float* __restrict__ gamma,   // [64]
    const float* __restrict__ beta,    // [64]
    const float* __restrict__ mean,    // [64]
    const float* __restrict__ var,     // [64]
    float* __restrict__ out)           // [8, 64, 496, 432]
{
    __shared__ v2f lds_bf[16 * 4 * 32];  // 16 KB, fragment-order weights

    // Build the swizzled fragment table straight from global (w is 16 KB and
    // L2-resident; each thread emits 16 fragments).
    for (int e = threadIdx.x; e < 16 * 4 * 32; e += 128) {
        const int l  = e & 31;
        const int nt = (e >> 5) & 3;
        const int ks = e >> 7;
        const int ko = (l >> 4) << 1;            // 0 or 2
        const int n  = nt * 16 + (l & 15);
        const int k  = ks * 4 + ko;
        v2f f;
        f.x = w[k * C_DIM + n];
        f.y = w[(k + 1) * C_DIM + n];
        lds_bf[e] = f;
    }
    __syncthreads();

    const int lane    = threadIdx.x & 31;
    const int waveId  = threadIdx.x >> 5;
    const int tile    = blockIdx.x * 4 + waveId;  // 8192 tiles total
    const int rowBase = tile * 16;
    const int m       = lane & 15;                // column-in-tile / row-in-A
    const int koff    = (lane >> 4) << 1;         // 0 for lanes 0-15, 2 for 16-31

    // Preload all A fragments for this wave's 16 rows (32 floats/lane).
    v2f a[16];
    const float* ap = pf + (size_t)(rowBase + m) * C_DIM + koff;
#pragma unroll
    for (int ks = 0; ks < 16; ++ks) {
        a[ks] = *(const v2f*)(ap + ks * 4);
    }

    v8f acc[4];
#pragma unroll
    for (int nt = 0; nt < 4; ++nt) acc[nt] = (v8f){0, 0, 0, 0, 0, 0, 0, 0};

    // K = 64 in 16 steps of 4; 4 N-tiles of 16 -> 64 WMMA ops per wave.
    // Each B fragment is one aligned, bank-conflict-free ds_load_b64.
    const v2f* bfp = lds_bf + lane;
#pragma unroll
    for (int ks = 0; ks < 16; ++ks) {
#pragma unroll
        for (int nt = 0; nt < 4; ++nt) {
            v2f bf = bfp[(ks * 4 + nt) * 32];
            acc[nt] = __builtin_amdgcn_wmma_f32_16x16x4_f32(
                /*neg_a=*/false, a[ks], /*neg_b=*/false, bf,
                /*c_mod=*/(short)0, acc[nt],
                /*reuse_a=*/false, /*reuse_b=*/false);
        }
    }

    // Per-channel BN fold: val = acc*s + t,  s = g*rsqrt(v+eps),
    //                      t = (bias - mean)*s + beta.   (channel = nt*16 + m)
    float s[4], t[4];
#pragma unroll
    for (int nt = 0; nt < 4; ++nt) {
        const int ch = nt * 16 + m;
        const float sc = gamma[ch] * __frsqrt_rn(var[ch] + BN_EPS);
        s[nt] = sc;
        t[nt] = (bias[ch] - mean[ch]) * sc + beta[ch];
    }

    // Scatter: VGPR v of acc holds row M = v (lanes 0-15) or v+8 (lanes 16-31).
    const int plane = NY_DIM * NX_DIM;  // 214272
#pragma unroll
    for (int v = 0; v < 8; ++v) {
        const int row = rowBase + v + ((lane >> 4) << 3);
        const int* cr = coords + (size_t)row * 5;
        const int b    = cr[0];
        const int x    = cr[1];
        const int y    = cr[2];
        const int flag = cr[4];
        if (flag != -1) {
            const int base = (b * C_DIM) * plane + y * NX_DIM + x;
#pragma unroll
            for (int nt = 0; nt < 4; ++nt) {
                const int ch = nt * 16 + m;
                float val = acc[nt][v] * s[nt] + t[nt];
                val = fmaxf(val, 0.f);
                out[base + ch * plane] = val;
            }
        }
    }
}

// ---------------------------------------------------------------------------
// Launch wrapper.
// Input order: pillar_features, voxel_coords, topk_w, topk_b,
//              bn_gamma, bn_beta, bn_mean, bn_var, batch_size, nx, ny
// ---------------------------------------------------------------------------
extern "C" void kernel_launch(void* const* d_in, const int* in_sizes, int n_in,
                              void* d_out, int out_size, void* d_ws, size_t ws_size,
                              hipStream_t stream) {
    const float* pf     = (const float*)d_in[0];
    const int*   coords = (const int*)d_in[1];
    const float* w      = (const float*)d_in[2];
    const float* bias   = (const float*)d_in[3];
    const float* gamma  = (const float*)d_in[4];
    const float* beta   = (const float*)d_in[5];
    const float* mean   = (const float*)d_in[6];
    const float* var    = (const float*)d_in[7];
    float* out = (float*)d_out;

    // Stage 1: zero the dense grid (out_size = 8*64*496*432, divisible by 4).
    const unsigned int n4 = (unsigned int)(out_size / 4);
    const int zblocks = 8192;
    pps_zero_kernel<<<zblocks, 256, 0, stream>>>((v4f*)out, n4);

    // Stage 2: fused WMMA GEMM + BN + ReLU + scatter.
    // 131072 pillars / 16 rows per wave = 8192 waves; 4 waves per block.
    const int gblocks = (P_TOTAL / 16) / 4;  // 2048
    pps_gemm_scatter_kernel<<<gblocks, 128, 0, stream>>>(
        pf, coords, w, bias, gamma, beta, mean, var, out);
}